// StaticGat_78675210928330
// MI455X (gfx1250) — compile-verified
//
#include <hip/hip_runtime.h>
#include <math.h>

typedef float v2f __attribute__((ext_vector_type(2)));
typedef float v8f __attribute__((ext_vector_type(8)));

#define HEADS 8
#define HDIM 16
#define IN_DIM 128
#define HID 128

// ---------------------------------------------------------------------------
// Init: m = -1e30, denom = 0, agg = 0 (both graphs)
// ---------------------------------------------------------------------------
__global__ void init_kernel(float* m_i, float* m_j, float* den_i, float* den_j,
                            float* agg_i, float* agg_j, int NH, int NZ) {
    int i = blockIdx.x * blockDim.x + threadIdx.x;
    if (i < NH) {
        m_i[i] = -1.0e30f; m_j[i] = -1.0e30f;
        den_i[i] = 0.0f;   den_j[i] = 0.0f;
    }
    if (i < NZ) {
        agg_i[i] = 0.0f;   agg_j[i] = 0.0f;
    }
}

// ---------------------------------------------------------------------------
// z = X @ W using V_WMMA_F32_16X16X4_F32.
// Block = 256 threads = 8 waves; wave w computes the 16x16 tile at
// (rowTile = blockIdx.x, colTile = w). W (128x128 f32 = 64KB) staged in LDS.
// A layout (16x4 f32): lane l -> M = l&15, comp c -> K = 2*(l>>4)+c.
// B layout (4x16 f32): lane l -> N = l&15, comp c -> K = 2*(l>>4)+c.
// C/D (16x16 f32):     lane l, vgpr v -> M = v + 8*(l>>4), N = l&15.
// ---------------------------------------------------------------------------
__global__ void gemm_wmma_kernel(const float* __restrict__ X,
                                 const float* __restrict__ W,
                                 float* __restrict__ Z, int n_rows) {
    __shared__ float lds_w[IN_DIM * HID];   // 64 KB (<= 320 KB WGP LDS)

    const int tid = threadIdx.x;
    for (int i = tid * 4; i < IN_DIM * HID; i += blockDim.x * 4) {
        *(float4*)(lds_w + i) = *(const float4*)(W + i);
    }
    __syncthreads();

    const int wave  = tid >> 5;          // 0..7  -> N tile
    const int lane  = tid & 31;
    const int mlo   = lane & 15;
    const int khalf = lane >> 4;         // 0/1
    const int row   = blockIdx.x * 16 + mlo;
    const int rclmp = row < n_rows ? row : (n_rows - 1);
    const int ncol  = wave * 16 + mlo;

    const float* xrow = X + rclmp * IN_DIM;

    v8f acc = {0.f, 0.f, 0.f, 0.f, 0.f, 0.f, 0.f, 0.f};
    for (int kk = 0; kk < IN_DIM; kk += 4) {
        const int ka = kk + 2 * khalf;
        v2f a = *(const v2f*)(xrow + ka);
        v2f b;
        b.x = lds_w[ka * HID + ncol];
        b.y = lds_w[(ka + 1) * HID + ncol];
        acc = __builtin_amdgcn_wmma_f32_16x16x4_f32(
            /*neg_a=*/false, a, /*neg_b=*/false, b,
            /*c_mod=*/(short)0, acc, /*reuse_a=*/false, /*reuse_b=*/false);
    }

    if (row < n_rows) {
        float* zbase = Z + (blockIdx.x * 16) * HID + wave * 16;
#pragma unroll
        for (int v = 0; v < 8; ++v) {
            int m = v + 8 * khalf;
            if (blockIdx.x * 16 + m < n_rows)
                zbase[m * HID + mlo] = acc[v];
        }
    }
}

// ---------------------------------------------------------------------------
// Per-(node, head) attention scores: e_s = <z[n,h,:], a_src[h,:]>, e_d likewise.
// ---------------------------------------------------------------------------
__global__ void scores_kernel(const float* __restrict__ Z,
                              const float* __restrict__ a_src,
                              const float* __restrict__ a_dst,
                              float* __restrict__ es, float* __restrict__ ed,
                              int n) {
    int i = blockIdx.x * blockDim.x + threadIdx.x;
    if (i >= n * HEADS) return;
    int node = i >> 3, h = i & 7;
    const float* zp = Z + node * HID + h * HDIM;
    const float* as = a_src + h * HDIM;
    const float* ad = a_dst + h * HDIM;
    float s = 0.f, d = 0.f;
#pragma unroll
    for (int t = 0; t < HDIM; ++t) {
        float zv = zp[t];
        s += zv * as[t];
        d += zv * ad[t];
    }
    es[i] = s;
    ed[i] = d;
}

__device__ __forceinline__ float leaky(float v) {
    return v > 0.f ? v : 0.2f * v;
}

// ---------------------------------------------------------------------------
// Edge pass 1: segment max via float atomicMax (global_atomic_max_num_f32)
// ---------------------------------------------------------------------------
__global__ void edge_max_kernel(const int* __restrict__ src, const int* __restrict__ dst,
                                const float* __restrict__ es, const float* __restrict__ ed,
                                float* __restrict__ m, int E) {
    int i = blockIdx.x * blockDim.x + threadIdx.x;
    if (i >= E * HEADS) return;
    int e = i >> 3, h = i & 7;
    int s = src[e], d = dst[e];
    float v = leaky(es[s * HEADS + h] + ed[d * HEADS + h]);
    atomicMax(&m[d * HEADS + h], v);
}

// ---------------------------------------------------------------------------
// Edge pass 2: denom = segment_sum(exp(e - m[dst]))
// ---------------------------------------------------------------------------
__global__ void edge_sum_kernel(const int* __restrict__ src, const int* __restrict__ dst,
                                const float* __restrict__ es, const float* __restrict__ ed,
                                const float* __restrict__ m, float* __restrict__ den, int E) {
    int i = blockIdx.x * blockDim.x + threadIdx.x;
    if (i >= E * HEADS) return;
    int e = i >> 3, h = i & 7;
    int s = src[e], d = dst[e];
    float v = leaky(es[s * HEADS + h] + ed[d * HEADS + h]);
    float w = __expf(v - m[d * HEADS + h]);
    atomicAdd(&den[d * HEADS + h], w);
}

// ---------------------------------------------------------------------------
// Edge pass 3: agg[dst] += alpha * z[src]  (16 float atomic adds per edge-head)
// ---------------------------------------------------------------------------
__global__ void edge_agg_kernel(const int* __restrict__ src, const int* __restrict__ dst,
                                const float* __restrict__ es, const float* __restrict__ ed,
                                const float* __restrict__ m, const float* __restrict__ den,
                                const float* __restrict__ Z, float* __restrict__ agg, int E) {
    int i = blockIdx.x * blockDim.x + threadIdx.x;
    if (i >= E * HEADS) return;
    int e = i >> 3, h = i & 7;
    int s = src[e], d = dst[e];
    float v = leaky(es[s * HEADS + h] + ed[d * HEADS + h]);
    float alpha = __expf(v - m[d * HEADS + h]) / den[d * HEADS + h];
    const float* zp = Z + s * HID + h * HDIM;
    float* ap = agg + d * HID + h * HDIM;
#pragma unroll
    for (int t = 0; t < HDIM; ++t) {
        atomicAdd(ap + t, alpha * zp[t]);
    }
}

// ---------------------------------------------------------------------------
// Output: one wave32 per batch element. L2-normalize via dot/||.||, then
// out[b] = <sage_i[idx_i[b]], W_out[:128]> + <sage_j[idx_j[b]], W_out[128:]> + b
// ---------------------------------------------------------------------------
__global__ void out_kernel(const float* __restrict__ agg_i, const float* __restrict__ agg_j,
                           const int* __restrict__ idx_i, const int* __restrict__ idx_j,
                           const float* __restrict__ W_out, const float* __restrict__ b_out,
                           float* __restrict__ out, int B) {
    int wid  = (blockIdx.x * blockDim.x + threadIdx.x) >> 5;
    int lane = threadIdx.x & 31;
    if (wid >= B) return;
    const float* pi = agg_i + idx_i[wid] * HID;
    const float* pj = agg_j + idx_j[wid] * HID;
    float ssi = 0.f, doti = 0.f, ssj = 0.f, dotj = 0.f;
#pragma unroll
    for (int t = lane; t < HID; t += 32) {
        float xi = pi[t]; ssi += xi * xi; doti += xi * W_out[t];
        float xj = pj[t]; ssj += xj * xj; dotj += xj * W_out[HID + t];
    }
#pragma unroll
    for (int off = 16; off > 0; off >>= 1) {
        ssi  += __shfl_xor(ssi,  off, 32);
        doti += __shfl_xor(doti, off, 32);
        ssj  += __shfl_xor(ssj,  off, 32);
        dotj += __shfl_xor(dotj, off, 32);
    }
    if (lane == 0) {
        float ni = fmaxf(sqrtf(ssi), 1e-12f);
        float nj = fmaxf(sqrtf(ssj), 1e-12f);
        out[wid] = doti / ni + dotj / nj + b_out[0];
    }
}

// ---------------------------------------------------------------------------
extern "C" void kernel_launch(void* const* d_in, const int* in_sizes, int n_in,
                              void* d_out, int out_size, void* d_ws, size_t ws_size,
                              hipStream_t stream) {
    // setup_inputs() order
    const float* x_i   = (const float*)d_in[1];
    const float* x_j   = (const float*)d_in[2];
    const int*   src_i = (const int*)d_in[3];
    const int*   dst_i = (const int*)d_in[4];
    const int*   src_j = (const int*)d_in[5];
    const int*   dst_j = (const int*)d_in[6];
    const int*   idx_i = (const int*)d_in[7];
    const int*   idx_j = (const int*)d_in[8];
    const float* W_gat = (const float*)d_in[9];
    const float* a_src = (const float*)d_in[10];
    const float* a_dst = (const float*)d_in[11];
    const float* W_out = (const float*)d_in[12];
    const float* b_out = (const float*)d_in[13];
    float* out = (float*)d_out;

    const int N  = in_sizes[1] / IN_DIM;   // 20000
    const int E  = in_sizes[3];            // 640000
    const int B  = in_sizes[7];            // 16384
    const int NZ = N * HID;                // z / agg size per graph
    const int NH = N * HEADS;

    // Workspace carve-up (floats)
    float* ws = (float*)d_ws;
    float* z_i   = ws;                float* z_j   = z_i + NZ;
    float* es_i  = z_j + NZ;          float* ed_i  = es_i + NH;
    float* es_j  = ed_i + NH;         float* ed_j  = es_j + NH;
    float* m_i   = ed_j + NH;         float* m_j   = m_i + NH;
    float* den_i = m_j + NH;          float* den_j = den_i + NH;
    float* agg_i = den_j + NH;        float* agg_j = agg_i + NZ;

    const int TB = 256;

    // 0) init
    init_kernel<<<(NZ + TB - 1) / TB, TB, 0, stream>>>(m_i, m_j, den_i, den_j,
                                                       agg_i, agg_j, NH, NZ);

    // 1) z = x @ W  (WMMA f32 16x16x4)
    const int mtiles = (N + 15) / 16;
    gemm_wmma_kernel<<<mtiles, TB, 0, stream>>>(x_i, W_gat, z_i, N);
    gemm_wmma_kernel<<<mtiles, TB, 0, stream>>>(x_j, W_gat, z_j, N);

    // 2) per-node attention scores
    const int nhBlocks = (NH + TB - 1) / TB;
    scores_kernel<<<nhBlocks, TB, 0, stream>>>(z_i, a_src, a_dst, es_i, ed_i, N);
    scores_kernel<<<nhBlocks, TB, 0, stream>>>(z_j, a_src, a_dst, es_j, ed_j, N);

    // 3) edge passes
    const int ehBlocks = (E * HEADS + TB - 1) / TB;
    edge_max_kernel<<<ehBlocks, TB, 0, stream>>>(src_i, dst_i, es_i, ed_i, m_i, E);
    edge_max_kernel<<<ehBlocks, TB, 0, stream>>>(src_j, dst_j, es_j, ed_j, m_j, E);
    edge_sum_kernel<<<ehBlocks, TB, 0, stream>>>(src_i, dst_i, es_i, ed_i, m_i, den_i, E);
    edge_sum_kernel<<<ehBlocks, TB, 0, stream>>>(src_j, dst_j, es_j, ed_j, m_j, den_j, E);
    edge_agg_kernel<<<ehBlocks, TB, 0, stream>>>(src_i, dst_i, es_i, ed_i, m_i, den_i, z_i, agg_i, E);
    edge_agg_kernel<<<ehBlocks, TB, 0, stream>>>(src_j, dst_j, es_j, ed_j, m_j, den_j, z_j, agg_j, E);

    // 4) gather + L2-normalized dot -> out[1,B]
    const int outBlocks = (B * 32 + TB - 1) / TB;
    out_kernel<<<outBlocks, TB, 0, stream>>>(agg_i, agg_j, idx_i, idx_j,
                                             W_out, b_out, out, B);
}